// proto_net_loss_35304631174173
// MI455X (gfx1250) — compile-verified
//
#include <hip/hip_runtime.h>
#include <hip/hip_bf16.h>

typedef __attribute__((ext_vector_type(2))) float v2f;
typedef __attribute__((ext_vector_type(8))) float v8f;

#define HID 128          // hidden/output feature dim
#define NX  8192         // variations rows
#define NY  2048         // exemplar rows
#define LAT 512          // latent dim
#define BPITCH 132       // LDS pitch (mod 64 == 4 -> conflict-free b64 reads)

// ---------------------------------------------------------------------------
// Kernel 1: H = X @ W1   (X: M x 512, W1: 512 x 128, H: M x 128)
// Block: 256 threads = 8 waves. Block handles 16 rows; wave w -> cols [16w,16w+16).
// f32 WMMA 16x16x4, K = 512 -> 128 WMMA per wave.
// A-frag layout (16x4 f32): lanes 0-15 = rows, VGPR0/1 = K{0,1}; lanes 16-31 = K{2,3}.
// ---------------------------------------------------------------------------
__global__ __launch_bounds__(256) void k_gemm1(const float* __restrict__ X,
                                               const float* __restrict__ W,
                                               float* __restrict__ H,
                                               int K) {
    const int wave = threadIdx.x >> 5;
    const int lane = threadIdx.x & 31;
    const int half = lane >> 4;
    const int idx  = lane & 15;
    const int row0 = blockIdx.x * 16;
    const int col0 = wave * 16;

    const float* __restrict__ Arow = X + (size_t)(row0 + idx) * K;

    v8f acc = {};
#pragma unroll 4
    for (int k = 0; k < K; k += 4) {
        const int ka = k + 2 * half;
        v2f a = *(const v2f*)(Arow + ka);                       // 8B-aligned (ka even)
        v2f b;
        b.x = W[(size_t)ka       * HID + col0 + idx];
        b.y = W[(size_t)(ka + 1) * HID + col0 + idx];
        acc = __builtin_amdgcn_wmma_f32_16x16x4_f32(
            false, a, false, b, (short)0, acc, false, false);
    }
#pragma unroll
    for (int v = 0; v < 8; ++v)
        H[(size_t)(row0 + v + 8 * half) * HID + col0 + idx] = acc[v];
}

// ---------------------------------------------------------------------------
// Kernel 2: per-column batch statistics -> folded BN affine (scale, shift)
// grid = 128 columns, 256 threads reduce over rows.
// ---------------------------------------------------------------------------
__global__ __launch_bounds__(256) void k_colstats(const float* __restrict__ H, int rows,
                                                  const float* __restrict__ gamma,
                                                  const float* __restrict__ beta,
                                                  float* __restrict__ scale,
                                                  float* __restrict__ shift) {
    __shared__ float s_sum[256];
    __shared__ float s_sq[256];
    const int col = blockIdx.x;
    float sum = 0.f, sq = 0.f;
    for (int r = threadIdx.x; r < rows; r += 256) {
        float v = H[(size_t)r * HID + col];
        sum += v;
        sq  = fmaf(v, v, sq);
    }
    s_sum[threadIdx.x] = sum;
    s_sq[threadIdx.x]  = sq;
    __syncthreads();
    for (int s = 128; s > 0; s >>= 1) {
        if (threadIdx.x < s) {
            s_sum[threadIdx.x] += s_sum[threadIdx.x + s];
            s_sq[threadIdx.x]  += s_sq[threadIdx.x + s];
        }
        __syncthreads();
    }
    if (threadIdx.x == 0) {
        const float inv_n = 1.0f / (float)rows;
        const float mu  = s_sum[0] * inv_n;
        const float var = s_sq[0] * inv_n - mu * mu;      // biased var (torch BN norm)
        const float sc  = gamma[col] * rsqrtf(var + 1e-5f);
        scale[col] = sc;
        shift[col] = beta[col] - mu * sc;
    }
}

// ---------------------------------------------------------------------------
// Kernel 3: Z = relu(H*scale + shift) @ W2   (K = N = 128), BN+ReLU fused into
// the A-fragment build. Same 8-wave / 16-row block structure as k_gemm1.
// ---------------------------------------------------------------------------
__global__ __launch_bounds__(256) void k_gemm2(const float* __restrict__ H,
                                               const float* __restrict__ W2,
                                               const float* __restrict__ scale,
                                               const float* __restrict__ shift,
                                               float* __restrict__ Z) {
    __shared__ float sS[HID];
    __shared__ float sH[HID];
    if (threadIdx.x < HID) {
        sS[threadIdx.x] = scale[threadIdx.x];
        sH[threadIdx.x] = shift[threadIdx.x];
    }
    __syncthreads();

    const int wave = threadIdx.x >> 5;
    const int lane = threadIdx.x & 31;
    const int half = lane >> 4;
    const int idx  = lane & 15;
    const int row0 = blockIdx.x * 16;
    const int col0 = wave * 16;

    const float* __restrict__ Arow = H + (size_t)(row0 + idx) * HID;

    v8f acc = {};
#pragma unroll 4
    for (int k = 0; k < HID; k += 4) {
        const int ka = k + 2 * half;
        v2f h2 = *(const v2f*)(Arow + ka);
        v2f a;
        a.x = fmaxf(fmaf(h2.x, sS[ka],     sH[ka]),     0.f);
        a.y = fmaxf(fmaf(h2.y, sS[ka + 1], sH[ka + 1]), 0.f);
        v2f b;
        b.x = W2[(size_t)ka       * HID + col0 + idx];
        b.y = W2[(size_t)(ka + 1) * HID + col0 + idx];
        acc = __builtin_amdgcn_wmma_f32_16x16x4_f32(
            false, a, false, b, (short)0, acc, false, false);
    }
#pragma unroll
    for (int v = 0; v < 8; ++v)
        Z[(size_t)(row0 + v + 8 * half) * HID + col0 + idx] = acc[v];
}

// ---------------------------------------------------------------------------
// Kernel 4: n2[row] = sum_k Z[row][k]^2 ; one 128-thread block per row.
// ---------------------------------------------------------------------------
__global__ __launch_bounds__(128) void k_rownorm(const float* __restrict__ Z,
                                                 float* __restrict__ n2) {
    __shared__ float s[128];
    const float v = Z[(size_t)blockIdx.x * HID + threadIdx.x];
    s[threadIdx.x] = v * v;
    __syncthreads();
    for (int st = 64; st > 0; st >>= 1) {
        if (threadIdx.x < st) s[threadIdx.x] += s[threadIdx.x + st];
        __syncthreads();
    }
    if (threadIdx.x == 0) n2[blockIdx.x] = s[0];
}

// ---------------------------------------------------------------------------
// Kernel 5: Out[i][j] = x2[i] + y2[j] - 2 * dot(Z1[i], Z2[j])
// Block: 256 threads = 8 waves -> 128x16 output tile. The 16x128 Z2 tile is
// staged in LDS (pitch 132 -> conflict-free ds_load_b64 across both
// half-waves). Each wave runs 32 f32 WMMAs (K=128).
// ---------------------------------------------------------------------------
__global__ __launch_bounds__(256) void k_pairwise(const float* __restrict__ Z1,
                                                  const float* __restrict__ Z2,
                                                  const float* __restrict__ X2,
                                                  const float* __restrict__ Y2,
                                                  float* __restrict__ Out) {
    __shared__ float sB[16 * BPITCH];
    const int row0 = blockIdx.x * 128;
    const int col0 = blockIdx.y * 16;

    // Stage B tile: 16 cols x 128 K values (8 KB)
    for (int t = threadIdx.x; t < 16 * HID; t += 256) {
        const int j = t >> 7;        // 0..15 (output column within tile)
        const int k = t & (HID - 1); // 0..127
        sB[j * BPITCH + k] = Z2[(size_t)(col0 + j) * HID + k];
    }
    __syncthreads();

    const int wave = threadIdx.x >> 5;
    const int lane = threadIdx.x & 31;
    const int half = lane >> 4;
    const int idx  = lane & 15;
    const int r0   = row0 + wave * 16;

    const float* __restrict__ Arow = Z1 + (size_t)(r0 + idx) * HID;
    const float* __restrict__ Brow = &sB[idx * BPITCH];

    v8f acc = {};
#pragma unroll 8
    for (int k = 0; k < HID; k += 4) {
        const int ka = k + 2 * half;
        v2f a = *(const v2f*)(Arow + ka);   // global_load_b64 (L2-resident)
        v2f b = *(const v2f*)(Brow + ka);   // ds_load_b64, bank-conflict-free
        acc = __builtin_amdgcn_wmma_f32_16x16x4_f32(
            false, a, false, b, (short)0, acc, false, false);
    }

    const float y2v = Y2[col0 + idx];
#pragma unroll
    for (int v = 0; v < 8; ++v) {
        const int r = r0 + v + 8 * half;
        Out[(size_t)r * NY + col0 + idx] = X2[r] + y2v - 2.0f * acc[v];
    }
}

// ---------------------------------------------------------------------------
extern "C" void kernel_launch(void* const* d_in, const int* in_sizes, int n_in,
                              void* d_out, int out_size, void* d_ws, size_t ws_size,
                              hipStream_t stream) {
    const float* X     = (const float*)d_in[0];  // (8192, 512)
    const float* Y     = (const float*)d_in[1];  // (2048, 512)
    const float* w1    = (const float*)d_in[2];  // (512, 128)
    const float* gamma = (const float*)d_in[3];  // (128,)
    const float* beta  = (const float*)d_in[4];  // (128,)
    const float* w2    = (const float*)d_in[5];  // (128, 128)
    float* out = (float*)d_out;                  // (8192, 2048)

    float* ws      = (float*)d_ws;
    float* h_x     = ws;                               // 8192*128
    float* h_y     = h_x + (size_t)NX * HID;           // 2048*128
    float* z_x     = h_y + (size_t)NY * HID;           // 8192*128
    float* z_y     = z_x + (size_t)NX * HID;           // 2048*128
    float* scale_x = z_y + (size_t)NY * HID;           // 128
    float* shift_x = scale_x + HID;                    // 128
    float* scale_y = shift_x + HID;                    // 128
    float* shift_y = scale_y + HID;                    // 128
    float* n2_x    = shift_y + HID;                    // 8192
    float* n2_y    = n2_x + NX;                        // 2048

    // Projection 1: hidden pre-activations
    k_gemm1<<<NX / 16, 256, 0, stream>>>(X, w1, h_x, LAT);
    k_gemm1<<<NY / 16, 256, 0, stream>>>(Y, w1, h_y, LAT);

    // Batch-norm statistics (train mode, per-set)
    k_colstats<<<HID, 256, 0, stream>>>(h_x, NX, gamma, beta, scale_x, shift_x);
    k_colstats<<<HID, 256, 0, stream>>>(h_y, NY, gamma, beta, scale_y, shift_y);

    // BN + ReLU fused into projection 2
    k_gemm2<<<NX / 16, 256, 0, stream>>>(h_x, w2, scale_x, shift_x, z_x);
    k_gemm2<<<NY / 16, 256, 0, stream>>>(h_y, w2, scale_y, shift_y, z_y);

    // Row squared norms
    k_rownorm<<<NX, 128, 0, stream>>>(z_x, n2_x);
    k_rownorm<<<NY, 128, 0, stream>>>(z_y, n2_y);

    // Pairwise squared distances
    k_pairwise<<<dim3(NX / 128, NY / 16), 256, 0, stream>>>(z_x, z_y, n2_x, n2_y, out);
}